// qms_nn_48799418417296
// MI455X (gfx1250) — compile-verified
//
#include <hip/hip_runtime.h>
#include <hip/hip_bf16.h>
#include <math.h>

// MI455X (gfx1250) implementation.
//
// Roofline: ~0.7 GFLOP vs ~270 MB of HBM traffic => memory bound (~12us floor
// at 23.3 TB/s). The 7 sequential batch-norms over B=2048 force a multi-kernel
// pipeline (batch stats span the grid). FP32 kept throughout.
//
// WMMA: gene einsum lin[b,g,h] = sum_f x[b,f*G+g] * gene_W[g,h,f] is computed
// as one V_WMMA_F32_16X16X4_F32 per (gene, 16-batch tile): A = 16x4 x-tile
// (M=batch, K=F), B = 4x16 gene_W^T (N=h, 6 of 16 cols used), K=4 exactly.
// NOTE: gene bias cancels in BN (tanh(bn(lin))) so both passes skip it.
//
// Assumed d_in layout (setup_inputs dict order, term_params flattened):
//   0 x, 1 gene_W, 2 gene_b(unused: cancels in BN), 3 gene_g, 4 gene_be,
//   5 gene_a1W, 6 gene_a1b, 7 gene_a2W, 8 gene_a2b,
//   9+8*l+{0..7}: level-l W,b,g,be,a1W,a1b,a2W,a2b  (l=0..4)
//   49 final_W, 50 final_b, 51 final_g, 52 final_be,
//   53 final_aW, 54 final_ab, 55 final_oW, 56 final_ob

typedef __attribute__((ext_vector_type(2))) float v2f;
typedef __attribute__((ext_vector_type(8))) float v8f;

#define NB   2048
#define NG   3000
#define NOUT 4250

// ---------------- workspace layout (float offsets) ----------------
// gene stats
#define GS   0u
#define GQ   18000u
#define GA   36000u
#define GBS  54000u
// level stats: sum, sumsq, scale(A), shift(B) per level, channels = T[l]*4
#define LS0  72000u
#define LQ0  76000u
#define LA0  80000u
#define LB0  84000u
#define LS1  88000u
#define LQ1  88800u
#define LA1  89600u
#define LB1  90400u
#define LS2  91200u
#define LQ2  91360u
#define LA2  91520u
#define LB2  91680u
#define LS3  91840u
#define LQ3  91872u
#define LA3  91904u
#define LB3  91936u
#define LS4  91968u
#define LQ4  91972u
#define LA4  91976u
#define LB4  91980u
#define FSO  91984u
#define FQO  91988u
#define FAO  91992u
#define FBO  91996u
#define STATS_FLOATS 92000u
// per-level tanh activation buffers [B][T[l]][4]
#define TB0  92000u
#define TB1  8284000u
#define TB2  9922400u
#define TB3  10250080u
#define TB4  10315616u
#define TFO  10323808u
// total = 10332000 floats = 41,328,000 bytes of d_ws

// ---------------- kernel 1: gene BN statistics (WMMA) ----------------
// grid (25, 16), block 128 (4 waves). Block covers 120 genes; each wave 30
// genes, looping 8 batch tiles of 16 (grid.y partitions batch into 16 x 128).
__global__ __launch_bounds__(128) void k_gene_stats(
    const float* __restrict__ x, const float* __restrict__ gW,
    float* __restrict__ sum, float* __restrict__ sumsq)
{
    const int lane = threadIdx.x & 31;
    const int wave = threadIdx.x >> 5;
    const int n    = lane & 15;        // A: M index, B: N index
    const int kb   = (lane >> 4) * 2;  // K base for this lane half
    const int gbase = blockIdx.x * 120 + wave * 30;

    for (int gi = 0; gi < 30; ++gi) {
        const int g = gbase + gi;
        // B matrix 4x16: B[k][n] = gene_W[g][n][k], n>=6 padded with 0
        v2f bm;
        bm.x = (n < 6) ? gW[g * 24 + n * 4 + kb]     : 0.0f;
        bm.y = (n < 6) ? gW[g * 24 + n * 4 + kb + 1] : 0.0f;

        float s = 0.0f, sq = 0.0f;
        for (int tile = 0; tile < 8; ++tile) {
            const int b0 = (blockIdx.y * 8 + tile) * 16;
            const float* xr = x + (size_t)(b0 + n) * 12000 + g;
            v2f a;
            a.x = xr[(size_t)kb * 3000];
            a.y = xr[(size_t)(kb + 1) * 3000];
            v8f c = {};
            c = __builtin_amdgcn_wmma_f32_16x16x4_f32(
                    false, a, false, bm, (short)0, c, false, false);
            #pragma unroll
            for (int r = 0; r < 8; ++r) { const float v = c[r]; s += v; sq += v * v; }
        }
        // lanes n and n+16 hold the two batch-halves of channel (g, n)
        s  += __shfl_xor(s, 16);
        sq += __shfl_xor(sq, 16);
        if (lane < 6) {
            atomicAdd(&sum[g * 6 + lane], s);
            atomicAdd(&sumsq[g * 6 + lane], sq);
        }
    }
}

// ---------------- finalize: fold BN into scale/shift ----------------
__global__ void k_finalize(const float* __restrict__ sum, const float* __restrict__ sumsq,
                           const float* __restrict__ gamma, const float* __restrict__ beta,
                           float* __restrict__ Aout, float* __restrict__ Bout, int nch)
{
    const int i = blockIdx.x * blockDim.x + threadIdx.x;
    if (i >= nch) return;
    const float mean = sum[i] * (1.0f / 2048.0f);
    const float var  = sumsq[i] * (1.0f / 2048.0f) - mean * mean;
    const float inv  = rsqrtf(fmaxf(var, 0.0f) + 1e-5f);
    const float A    = inv * gamma[i];
    Aout[i] = A;
    Bout[i] = beta[i] - mean * A;
}

// ---- kernel 3: gene_out (WMMA) + gene aux + term-0 forward + stats ----
// grid (25, 128): 120 genes x one 16-batch tile per block, block 128 (4 waves).
__global__ __launch_bounds__(128) void k_gene_term0(
    const float* __restrict__ x, const float* __restrict__ gW,
    const float* __restrict__ geneA, const float* __restrict__ geneSh,
    const float* __restrict__ a1W, const float* __restrict__ a1b,
    const float* __restrict__ a2W, const float* __restrict__ a2b,
    const float* __restrict__ W0, const float* __restrict__ bias0,
    float* __restrict__ outp, float* __restrict__ t0buf,
    float* __restrict__ sum0, float* __restrict__ sumsq0)
{
    __shared__ float sg[16][120][6];   // gene_out tile: 46 KB of 320 KB LDS
    __shared__ float sstat[320];       // term0 block-local stats (160 ch x 2)

    const int lane = threadIdx.x & 31;
    const int wave = threadIdx.x >> 5;
    const int n    = lane & 15;
    const int half = lane >> 4;
    const int kb   = half * 2;
    const int b0   = blockIdx.y * 16;
    const int gblk = blockIdx.x * 120;

    for (int i = threadIdx.x; i < 320; i += 128) sstat[i] = 0.0f;

    const int gbase = gblk + wave * 30;
    for (int gi = 0; gi < 30; ++gi) {
        const int g = gbase + gi;
        v2f bm;
        bm.x = (n < 6) ? gW[g * 24 + n * 4 + kb]     : 0.0f;
        bm.y = (n < 6) ? gW[g * 24 + n * 4 + kb + 1] : 0.0f;
        const float* xr = x + (size_t)(b0 + n) * 12000 + g;
        v2f a;
        a.x = xr[(size_t)kb * 3000];
        a.y = xr[(size_t)(kb + 1) * 3000];
        v8f c = {};
        c = __builtin_amdgcn_wmma_f32_16x16x4_f32(
                false, a, false, bm, (short)0, c, false, false);
        if (n < 6) {
            const float A_ = geneA[g * 6 + n];
            const float S_ = geneSh[g * 6 + n];
            const int   lg = wave * 30 + gi;
            #pragma unroll
            for (int r = 0; r < 8; ++r) {
                const int m = half ? (r + 8) : r;   // C layout: VGPR r -> M=r / M=r+8
                sg[m][lg][n] = tanhf(c[r] * A_ + S_);
            }
        }
    }
    __syncthreads();

    // gene aux head: out[b, 1+g]
    for (int idx = threadIdx.x; idx < 16 * 120; idx += 128) {
        const int m = idx / 120, lg = idx % 120;
        const int g = gblk + lg;
        float acc = a1b[g];
        #pragma unroll
        for (int h = 0; h < 6; ++h) acc += sg[m][lg][h] * a1W[g * 6 + h];
        const float fa = tanhf(acc);
        outp[(size_t)(b0 + m) * NOUT + 1 + g] = fa * a2W[g] + a2b[g];
    }

    // term level 0: 40 terms per block, in = 18 (3 genes x 6), out = 4
    for (int idx = threadIdx.x; idx < 2560; idx += 128) {
        const int o  = idx & 3;
        const int tl = (idx >> 2) % 40;
        const int m  = idx / 160;
        const int t  = blockIdx.x * 40 + tl;
        float acc = bias0[t * 4 + o];
        const float* w = W0 + (size_t)t * 72 + o * 18;
        #pragma unroll
        for (int i = 0; i < 18; ++i)
            acc += sg[m][tl * 3 + i / 6][i % 6] * w[i];
        const float th = tanhf(acc);
        t0buf[(size_t)(b0 + m) * 4000 + t * 4 + o] = th;
        atomicAdd(&sstat[(tl * 4 + o) * 2],     th);        // ds_add_f32
        atomicAdd(&sstat[(tl * 4 + o) * 2 + 1], th * th);
    }
    __syncthreads();
    for (int idx = threadIdx.x; idx < 160; idx += 128) {
        atomicAdd(&sum0[blockIdx.x * 160 + idx],   sstat[idx * 2]);
        atomicAdd(&sumsq0[blockIdx.x * 160 + idx], sstat[idx * 2 + 1]);
    }
}

// ---------------- generic term forward (levels 1..4) ----------------
// thread = (term, 16-batch chunk); normalizes prev on the fly via scale/shift.
__global__ void k_term_fwd(const float* __restrict__ prevBuf,
                           const float* __restrict__ prevA, const float* __restrict__ prevB,
                           const float* __restrict__ W, const float* __restrict__ bias,
                           float* __restrict__ outBuf,
                           float* __restrict__ sum, float* __restrict__ sumsq,
                           int Tprev, int Tcur, int IN)
{
    const int idx   = blockIdx.x * blockDim.x + threadIdx.x;
    const int total = Tcur * (NB / 16);
    if (idx >= total) return;
    const int t  = idx % Tcur;
    const int bc = idx / Tcur;

    float s[4] = {0, 0, 0, 0}, sq[4] = {0, 0, 0, 0};
    const float* w = W + (size_t)t * 4 * IN;
    for (int mb = 0; mb < 16; ++mb) {
        const int b = bc * 16 + mb;
        const float* pr = prevBuf + (size_t)b * Tprev * 4 + (size_t)t * IN;
        float acc[4];
        #pragma unroll
        for (int o = 0; o < 4; ++o) acc[o] = bias[t * 4 + o];
        for (int i = 0; i < IN; ++i) {
            const float v = pr[i] * prevA[t * IN + i] + prevB[t * IN + i];
            #pragma unroll
            for (int o = 0; o < 4; ++o) acc[o] += v * w[o * IN + i];
        }
        #pragma unroll
        for (int o = 0; o < 4; ++o) {
            const float th = tanhf(acc[o]);
            outBuf[(size_t)b * Tcur * 4 + t * 4 + o] = th;
            s[o] += th; sq[o] += th * th;
        }
    }
    #pragma unroll
    for (int o = 0; o < 4; ++o) {
        atomicAdd(&sum[t * 4 + o],   s[o]);
        atomicAdd(&sumsq[t * 4 + o], sq[o]);
    }
}

// ---------------- generic term aux head (levels 0..4) ----------------
__global__ void k_term_aux(const float* __restrict__ buf,
                           const float* __restrict__ A_, const float* __restrict__ B_,
                           const float* __restrict__ a1W, const float* __restrict__ a1b,
                           const float* __restrict__ a2W, const float* __restrict__ a2b,
                           float* __restrict__ outp, int T, int colBase)
{
    const int idx = blockIdx.x * blockDim.x + threadIdx.x;
    if (idx >= NB * T) return;
    const int t = idx % T, b = idx / T;
    const float* p = buf + (size_t)b * T * 4 + t * 4;
    float acc = a1b[t];
    #pragma unroll
    for (int o = 0; o < 4; ++o)
        acc += (p[o] * A_[t * 4 + o] + B_[t * 4 + o]) * a1W[t * 4 + o];
    const float fa = tanhf(acc);
    outp[(size_t)b * NOUT + colBase + t] = fa * a2W[t] + a2b[t];
}

// ---------------- final head forward: root -> tanh(linear) + stats ----------
__global__ void k_final_fwd(const float* __restrict__ t4buf,
                            const float* __restrict__ A4, const float* __restrict__ B4,
                            const float* __restrict__ fW, const float* __restrict__ fb,
                            float* __restrict__ fbuf,
                            float* __restrict__ sum, float* __restrict__ sumsq)
{
    const int b = blockIdx.x * blockDim.x + threadIdx.x;
    if (b >= NB) return;
    const int lane = threadIdx.x & 31;
    float root[4];
    #pragma unroll
    for (int j = 0; j < 4; ++j) root[j] = t4buf[b * 4 + j] * A4[j] + B4[j];
    #pragma unroll
    for (int o = 0; o < 4; ++o) {
        float acc = fb[o];
        #pragma unroll
        for (int j = 0; j < 4; ++j) acc += root[j] * fW[o * 4 + j];
        const float th = tanhf(acc);
        fbuf[b * 4 + o] = th;
        float vs = th, vq = th * th;
        #pragma unroll
        for (int off = 16; off > 0; off >>= 1) {
            vs += __shfl_down(vs, off);
            vq += __shfl_down(vq, off);
        }
        if (lane == 0) { atomicAdd(&sum[o], vs); atomicAdd(&sumsq[o], vq); }
    }
}

// ---------------- final aux: out[:, 0] ----------------
__global__ void k_final_aux(const float* __restrict__ fbuf,
                            const float* __restrict__ FA, const float* __restrict__ FB,
                            const float* __restrict__ faW, const float* __restrict__ fab,
                            const float* __restrict__ foW, const float* __restrict__ fob,
                            float* __restrict__ outp)
{
    const int b = blockIdx.x * blockDim.x + threadIdx.x;
    if (b >= NB) return;
    float acc = fab[0];
    #pragma unroll
    for (int o = 0; o < 4; ++o)
        acc += (fbuf[b * 4 + o] * FA[o] + FB[o]) * faW[o];
    const float fa = tanhf(acc);
    outp[(size_t)b * NOUT] = fa * foW[0] + fob[0];
}

extern "C" void kernel_launch(void* const* d_in, const int* in_sizes, int n_in,
                              void* d_out, int out_size, void* d_ws, size_t ws_size,
                              hipStream_t stream)
{
    const float* x    = (const float*)d_in[0];
    const float* gW   = (const float*)d_in[1];
    const float* gG   = (const float*)d_in[3];
    const float* gBe  = (const float*)d_in[4];
    const float* ga1W = (const float*)d_in[5];
    const float* ga1b = (const float*)d_in[6];
    const float* ga2W = (const float*)d_in[7];
    const float* ga2b = (const float*)d_in[8];
    auto tp = [&](int l, int k) { return (const float*)d_in[9 + l * 8 + k]; };
    const float* fW  = (const float*)d_in[49];
    const float* fb  = (const float*)d_in[50];
    const float* fG  = (const float*)d_in[51];
    const float* fBe = (const float*)d_in[52];
    const float* faW = (const float*)d_in[53];
    const float* fab = (const float*)d_in[54];
    const float* foW = (const float*)d_in[55];
    const float* fob = (const float*)d_in[56];

    float* outp = (float*)d_out;
    float* ws   = (float*)d_ws;

    const int    T_[5]   = {1000, 200, 40, 8, 1};
    const int    IN_[5]  = {18, 20, 20, 20, 32};
    const int    COL_[5] = {3001, 4001, 4201, 4241, 4249};
    const size_t LS_[5]  = {LS0, LS1, LS2, LS3, LS4};
    const size_t LQ_[5]  = {LQ0, LQ1, LQ2, LQ3, LQ4};
    const size_t LA_[5]  = {LA0, LA1, LA2, LA3, LA4};
    const size_t LB_[5]  = {LB0, LB1, LB2, LB3, LB4};
    const size_t TBuf[5] = {TB0, TB1, TB2, TB3, TB4};

    // zero all atomic-accumulated statistics (every launch: deterministic)
    hipMemsetAsync(d_ws, 0, STATS_FLOATS * sizeof(float), stream);

    // 1) gene BN stats via WMMA
    k_gene_stats<<<dim3(25, 16), 128, 0, stream>>>(x, gW, ws + GS, ws + GQ);
    k_finalize<<<(18000 + 255) / 256, 256, 0, stream>>>(
        ws + GS, ws + GQ, gG, gBe, ws + GA, ws + GBS, 18000);

    // 2) gene_out (WMMA) + gene aux + term0 forward + term0 stats
    k_gene_term0<<<dim3(25, 128), 128, 0, stream>>>(
        x, gW, ws + GA, ws + GBS, ga1W, ga1b, ga2W, ga2b,
        tp(0, 0), tp(0, 1), outp, ws + TBuf[0], ws + LS_[0], ws + LQ_[0]);

    // 3) term hierarchy
    for (int l = 0; l < 5; ++l) {
        const int nch = T_[l] * 4;
        k_finalize<<<(nch + 255) / 256, 256, 0, stream>>>(
            ws + LS_[l], ws + LQ_[l], tp(l, 2), tp(l, 3), ws + LA_[l], ws + LB_[l], nch);
        k_term_aux<<<(NB * T_[l] + 255) / 256, 256, 0, stream>>>(
            ws + TBuf[l], ws + LA_[l], ws + LB_[l],
            tp(l, 4), tp(l, 5), tp(l, 6), tp(l, 7), outp, T_[l], COL_[l]);
        if (l < 4) {
            const int total = T_[l + 1] * (NB / 16);
            k_term_fwd<<<(total + 255) / 256, 256, 0, stream>>>(
                ws + TBuf[l], ws + LA_[l], ws + LB_[l],
                tp(l + 1, 0), tp(l + 1, 1),
                ws + TBuf[l + 1], ws + LS_[l + 1], ws + LQ_[l + 1],
                T_[l], T_[l + 1], IN_[l + 1]);
        }
    }

    // 4) final head
    k_final_fwd<<<NB / 256, 256, 0, stream>>>(
        ws + TBuf[4], ws + LA_[4], ws + LB_[4], fW, fb,
        ws + TFO, ws + FSO, ws + FQO);
    k_finalize<<<1, 256, 0, stream>>>(ws + FSO, ws + FQO, fG, fBe, ws + FAO, ws + FBO, 4);
    k_final_aux<<<NB / 256, 256, 0, stream>>>(
        ws + TFO, ws + FAO, ws + FBO, faW, fab, foW, fob, outp);
}